// S4Block_48146583388754
// MI455X (gfx1250) — compile-verified
//
#include <hip/hip_runtime.h>
#include <hip/hip_bf16.h>
#include <math.h>

// Problem constants (match reference)
#define Bdim   8
#define Hdim   32
#define Ldim   32768
#define Ndim   32
#define CDdim  128
#define NCHUNK 32                 // chunks per sequence (must be 32: code uses >>5)
#define CHUNK  (Ldim / NCHUNK)    // 1024

typedef __attribute__((ext_vector_type(2))) float v2f;
typedef __attribute__((ext_vector_type(8))) float v8f;
typedef __attribute__((ext_vector_type(4))) int   i32x4;
typedef __attribute__((ext_vector_type(8))) int   i32x8;

// Branch-free gelu (tanh approximation): 0.5x(1+tanh(z)) == x*sigmoid(2z)
__device__ __forceinline__ float gelu_t(float x) {
    float x3 = x * x * x;
    float z  = fmaf(0.044715f, x3, x) * 1.5957691216057308f;  // 2*sqrt(2/pi)
    return x * __fdividef(1.0f, 1.0f + __expf(-z));
}
__device__ __forceinline__ float sigm(float x) {
    return __fdividef(1.0f, 1.0f + __expf(-x));
}
// Wave-uniform broadcast: v_readlane_b32 -> SGPR operand (beats ds_bpermute)
__device__ __forceinline__ float bcast(float v, int l) {
    return __uint_as_float(__builtin_amdgcn_readlane(__float_as_uint(v), l));
}
__device__ __forceinline__ float q_at(const float4& q, int comp) {
    return comp == 0 ? q.x : comp == 1 ? q.y : comp == 2 ? q.z : q.w;
}

// ---------------------------------------------------------------------------
// Tensor Data Mover: 2D f32 tile Global -> LDS (D# built in SGPR tuples).
// Group0: count=1 | lds_addr | global_addr(57b) | type=2.
// Group1: data_size=4B, tensor_dim0/1, tile_dim0/1, tensor_dim0_stride.
// ---------------------------------------------------------------------------
__device__ __forceinline__ void tdm_load_2d_f32(const void* gsrc, unsigned lds_off,
                                                int width, int height, int stride) {
    unsigned long long ga = (unsigned long long)(uintptr_t)gsrc;
    i32x4 g0;
    g0.x = 1;                                                   // count=1, user D#
    g0.y = (int)lds_off;                                        // lds_addr [63:32]
    g0.z = (int)(unsigned)ga;                                   // global_addr [95:64]
    g0.w = (int)(((unsigned)(ga >> 32) & 0x01FFFFFFu) | 0x80000000u); // [120:96] | type=2
    i32x8 g1;
    g1[0] = 2 << 16;                                            // data_size = 4 bytes
    g1[1] = (width & 0xFFFF) << 16;                             // tensor_dim0 lo16 @bit48
    g1[2] = ((width >> 16) & 0xFFFF) | ((height & 0xFFFF) << 16); // dim0 hi | dim1 lo
    g1[3] = ((height >> 16) & 0xFFFF) | ((width & 0xFFFF) << 16); // dim1 hi | tile_dim0
    g1[4] = (height & 0xFFFF);                                  // tile_dim1 (tile_dim2=0)
    g1[5] = stride;                                             // tensor_dim0_stride lo32
    g1[6] = 0;
    g1[7] = 0;
    asm volatile("tensor_load_to_lds %0, %1" :: "s"(g0), "s"(g1) : "memory");
}

// ---------------------------------------------------------------------------
// K0: per-(h,n) SSM constants: w = exp(dt*A), Cd = 2*C*(exp(dtA)-1)/A, Wc = w^CHUNK
// ---------------------------------------------------------------------------
__global__ void k_setup(const float* __restrict__ log_dt,
                        const float* __restrict__ A_re, const float* __restrict__ A_im,
                        const float* __restrict__ C_re, const float* __restrict__ C_im,
                        float* wr_, float* wi_, float* cdr_, float* cdi_,
                        float* wcr_, float* wci_) {
    int i = threadIdx.x;                 // H*N = 1024
    int h = i >> 5;
    float dt  = __expf(log_dt[h]);
    float ar  = A_re[i], ai = A_im[i];
    float dre = dt * ar, dim = dt * ai;
    float er  = __expf(dre);
    float sn, cs; __sincosf(dim, &sn, &cs);
    float wr = er * cs, wi = er * sn;
    float exr = wr - 1.0f, exi = wi;
    float den = ar * ar + ai * ai;
    float qr = (exr * ar + exi * ai) / den;
    float qi = (exi * ar - exr * ai) / den;
    float cr = C_re[i], ci = C_im[i];
    wr_[i]  = wr;  wi_[i]  = wi;
    cdr_[i] = 2.0f * (cr * qr - ci * qi);
    cdi_[i] = 2.0f * (cr * qi + ci * qr);
    float er2 = __expf(dre * (float)CHUNK);
    float sn2, cs2; __sincosf(dim * (float)CHUNK, &sn2, &cs2);
    wcr_[i] = er2 * cs2;  wci_[i] = er2 * sn2;
}

// ---------------------------------------------------------------------------
// K-film: gb[b,g] = cond[b] . film_W[g] + film_b[g]      (8 x 64 outputs)
// ---------------------------------------------------------------------------
__global__ void k_film(const float* __restrict__ cond, const float* __restrict__ fW,
                       const float* __restrict__ fb, float* __restrict__ gb) {
    int i = threadIdx.x;                 // 512
    int b = i >> 6;
    int g = i & 63;
    const float* c = cond + b * CDdim;
    const float* w = fW + g * CDdim;
    float s = fb[g];
    #pragma unroll 8
    for (int j = 0; j < CDdim; ++j) s = fmaf(c[j], w[j], s);
    gb[i] = s;
}

// ---------------------------------------------------------------------------
// K1: u = gelu(W_lin @ x + b_lin)   via V_WMMA_F32_16X16X4_F32
//     W_lin staged to LDS once per block through the TDM.
// ---------------------------------------------------------------------------
__global__ __launch_bounds__(256) void k_in_linear(
        const float* __restrict__ x, const float* __restrict__ W,
        const float* __restrict__ bias, float* __restrict__ u) {
    __shared__ __align__(16) float Wl[Hdim * Hdim];            // 4 KB
    if (threadIdx.x < 32) {
        tdm_load_2d_f32(W, (unsigned)(uintptr_t)Wl, Hdim, Hdim, Hdim);
        __builtin_amdgcn_s_wait_tensorcnt(0);
    }
    __syncthreads();

    const int lane = threadIdx.x & 31;
    const int m    = lane & 15;
    const int hi   = lane >> 4;
    const int gid  = blockIdx.x * 8 + (threadIdx.x >> 5);
    const int b    = gid / (Ldim / 16);
    const int l0   = (gid % (Ldim / 16)) * 16;
    const float* xb = x + (size_t)b * Hdim * Ldim;
    v8f acc0 = {};
    v8f acc1 = {};
    #pragma unroll
    for (int kk = 0; kk < 8; ++kk) {
        const int k = kk * 4 + 2 * hi;                         // K-pair for half-wave
        v2f bm;                                                // B: 4x16
        bm.x = xb[(size_t)k * Ldim + l0 + m];
        bm.y = xb[(size_t)(k + 1) * Ldim + l0 + m];
        v2f a0 = *(const v2f*)&Wl[m * Hdim + k];               // ds_load_b64
        v2f a1 = *(const v2f*)&Wl[(16 + m) * Hdim + k];
        acc0 = __builtin_amdgcn_wmma_f32_16x16x4_f32(false, a0, false, bm, (short)0, acc0, false, false);
        acc1 = __builtin_amdgcn_wmma_f32_16x16x4_f32(false, a1, false, bm, (short)0, acc1, false, false);
    }
    #pragma unroll
    for (int r = 0; r < 8; ++r) {
        int g0 = r + 8 * hi;
        int g1 = 16 + g0;
        u[((size_t)b * Hdim + g0) * Ldim + l0 + m] = gelu_t(acc0[r] + bias[g0]);
        u[((size_t)b * Hdim + g1) * Ldim + l0 + m] = gelu_t(acc1[r] + bias[g1]);
    }
}

// ---------------------------------------------------------------------------
// K2: local chunk scan (states only).  wave = (b,h,chunk), lane = state n.
//     u loaded 128 elems/iter (float4/lane), broadcast via v_readlane.
// ---------------------------------------------------------------------------
__global__ __launch_bounds__(256) void k_scan_local(
        const float* __restrict__ u, const float* __restrict__ wr_,
        const float* __restrict__ wi_, float* __restrict__ S) {
    int lane = threadIdx.x & 31;
    int gid  = blockIdx.x * 8 + (threadIdx.x >> 5);  // (b*H+h)*NCHUNK + c
    int c    = gid & (NCHUNK - 1);
    int bh   = gid >> 5;
    int h    = bh & (Hdim - 1);
    float wr = wr_[h * Ndim + lane], wi = wi_[h * Ndim + lane];
    const float* up = u + (size_t)bh * Ldim + (size_t)c * CHUNK;
    float sre = 0.0f, sim = 0.0f;
    for (int i = 0; i < CHUNK; i += 128) {
        float4 q = *(const float4*)(up + i + lane * 4);
        __builtin_prefetch(up + i + 512, 0, 1);      // global_prefetch_b8
        #pragma unroll
        for (int j = 0; j < 128; ++j) {
            float uv = bcast(q_at(q, j & 3), j >> 2);
            float nr = fmaf(wr, sre, fmaf(-wi, sim, uv));
            sim = fmaf(wi, sre, wr * sim);
            sre = nr;
        }
    }
    S[(size_t)gid * 64 + lane * 2]     = sre;
    S[(size_t)gid * 64 + lane * 2 + 1] = sim;
}

// ---------------------------------------------------------------------------
// K3: exclusive prefix over chunks (in place): P[c] = Wc*P[c-1] + S[c-1]
// ---------------------------------------------------------------------------
__global__ void k_scan_prefix(float* __restrict__ S,
                              const float* __restrict__ wcr_,
                              const float* __restrict__ wci_) {
    int t  = blockIdx.x * blockDim.x + threadIdx.x;  // B*H*N = 8192
    int n  = t & 31;
    int bh = t >> 5;
    int h  = bh & (Hdim - 1);
    float wr = wcr_[h * Ndim + n], wi = wci_[h * Ndim + n];
    float pr = 0.0f, pi = 0.0f;
    size_t base = (size_t)bh * NCHUNK * 64 + n * 2;
    for (int c = 0; c < NCHUNK; ++c) {
        size_t ix = base + (size_t)c * 64;
        float tr = S[ix], ti = S[ix + 1];
        S[ix] = pr;  S[ix + 1] = pi;                 // exclusive prefix
        float npr = fmaf(wr, pr, fmaf(-wi, pi, tr));
        pi = fmaf(wi, pr, fmaf(wr, pi, ti));
        pr = npr;
    }
}

// ---------------------------------------------------------------------------
// K4: output pass. Scan from prefix state; per step butterfly-reduce the
//     state contributions across lanes; D*u folded before the lane-select.
// ---------------------------------------------------------------------------
__global__ __launch_bounds__(256) void k_scan_out(
        const float* __restrict__ u, const float* __restrict__ S,
        const float* __restrict__ wr_, const float* __restrict__ wi_,
        const float* __restrict__ cdr_, const float* __restrict__ cdi_,
        const float* __restrict__ Dvec, float* __restrict__ v) {
    int lane = threadIdx.x & 31;
    int gid  = blockIdx.x * 8 + (threadIdx.x >> 5);
    int c    = gid & (NCHUNK - 1);
    int bh   = gid >> 5;
    int h    = bh & (Hdim - 1);
    float wr  = wr_[h * Ndim + lane],  wi  = wi_[h * Ndim + lane];
    float cdr = cdr_[h * Ndim + lane], cdi = cdi_[h * Ndim + lane];
    float Dh  = Dvec[h];
    float sre = S[(size_t)gid * 64 + lane * 2];
    float sim = S[(size_t)gid * 64 + lane * 2 + 1];
    const float* up = u + (size_t)bh * Ldim + (size_t)c * CHUNK;
    float*       vp = v + (size_t)bh * Ldim + (size_t)c * CHUNK;
    for (int i = 0; i < CHUNK; i += 128) {
        float4 q = *(const float4*)(up + i + lane * 4);
        __builtin_prefetch(up + i + 512, 0, 1);
        #pragma unroll
        for (int s = 0; s < 4; ++s) {
            float yreg = 0.0f;
            #pragma unroll
            for (int t = 0; t < 32; ++t) {
                const int j = s * 32 + t;
                float uv = bcast(q_at(q, j & 3), j >> 2);
                float nr = fmaf(wr, sre, fmaf(-wi, sim, uv));
                sim = fmaf(wi, sre, wr * sim);
                sre = nr;
                float tt = fmaf(cdr, sre, -(cdi * sim));
                #pragma unroll
                for (int off = 16; off; off >>= 1) tt += __shfl_xor(tt, off, 32);
                if (lane == t) yreg = fmaf(Dh, uv, tt);
            }
            vp[i + s * 32 + lane] = yreg;
        }
    }
}

// ---------------------------------------------------------------------------
// K5: z = W_out @ y + b_out ; GLU.  W_out (64x32) staged to LDS via TDM.
// ---------------------------------------------------------------------------
__global__ __launch_bounds__(256) void k_out_glu(
        const float* __restrict__ vin, const float* __restrict__ Wo,
        const float* __restrict__ bo, float* __restrict__ yg) {
    __shared__ __align__(16) float Wol[2 * Hdim * Hdim];       // 8 KB
    if (threadIdx.x < 32) {
        tdm_load_2d_f32(Wo, (unsigned)(uintptr_t)Wol, Hdim, 2 * Hdim, Hdim);
        __builtin_amdgcn_s_wait_tensorcnt(0);
    }
    __syncthreads();

    const int lane = threadIdx.x & 31;
    const int m    = lane & 15;
    const int hi   = lane >> 4;
    const int gid  = blockIdx.x * 8 + (threadIdx.x >> 5);
    const int b    = gid / (Ldim / 16);
    const int l0   = (gid % (Ldim / 16)) * 16;
    const float* vb = vin + (size_t)b * Hdim * Ldim;
    v8f acc[4] = {};
    #pragma unroll
    for (int kk = 0; kk < 8; ++kk) {
        const int k = kk * 4 + 2 * hi;
        v2f bm;
        bm.x = vb[(size_t)k * Ldim + l0 + m];
        bm.y = vb[(size_t)(k + 1) * Ldim + l0 + m];
        #pragma unroll
        for (int t = 0; t < 4; ++t) {
            v2f a = *(const v2f*)&Wol[(t * 16 + m) * Hdim + k];
            acc[t] = __builtin_amdgcn_wmma_f32_16x16x4_f32(false, a, false, bm, (short)0, acc[t], false, false);
        }
    }
    #pragma unroll
    for (int t = 0; t < 2; ++t) {
        #pragma unroll
        for (int r = 0; r < 8; ++r) {
            int g  = t * 16 + r + 8 * hi;
            float zv = acc[t][r]     + bo[g];
            float zg = acc[t + 2][r] + bo[g + Hdim];
            yg[((size_t)b * Hdim + g) * Ldim + l0 + m] = zv * sigm(zg);
        }
    }
}

// ---------------------------------------------------------------------------
// K6: per-channel mean / rstd over (B, L)  (deterministic block reduction)
// ---------------------------------------------------------------------------
__global__ __launch_bounds__(1024) void k_bn_stats(
        const float* __restrict__ yg, float* __restrict__ stats) {
    __shared__ float ls[1024];
    __shared__ float ls2[1024];
    int h = blockIdx.x;
    int tid = threadIdx.x;
    float s = 0.0f, s2 = 0.0f;
    for (int b = 0; b < Bdim; ++b) {
        const float* p = yg + ((size_t)b * Hdim + h) * Ldim;
        for (int i = tid; i < Ldim; i += 1024) {
            float val = p[i];
            s += val;  s2 = fmaf(val, val, s2);
        }
    }
    ls[tid] = s;  ls2[tid] = s2;
    __syncthreads();
    for (int st = 512; st > 0; st >>= 1) {
        if (tid < st) { ls[tid] += ls[tid + st]; ls2[tid] += ls2[tid + st]; }
        __syncthreads();
    }
    if (tid == 0) {
        float inv  = 1.0f / (float)((size_t)Bdim * Ldim);
        float mean = ls[0] * inv;
        float var  = ls2[0] * inv - mean * mean;
        stats[2 * h]     = mean;
        stats[2 * h + 1] = rsqrtf(var + 1e-5f);
    }
}

// ---------------------------------------------------------------------------
// K7: BN apply + FiLM + gelu + depthwise residual
// ---------------------------------------------------------------------------
__global__ __launch_bounds__(256) void k_final(
        const float* __restrict__ yg, const float* __restrict__ stats,
        const float* __restrict__ gb, const float* __restrict__ resw,
        const float* __restrict__ x, float* __restrict__ out) {
    size_t idx = (size_t)blockIdx.x * blockDim.x + threadIdx.x;
    int h = (int)((idx / Ldim) & (Hdim - 1));
    int b = (int)(idx / ((size_t)Hdim * Ldim));
    float yv    = yg[idx];
    float mean  = stats[2 * h];
    float rstd  = stats[2 * h + 1];
    float gamma = gb[b * (2 * Hdim) + h];
    float beta  = gb[b * (2 * Hdim) + Hdim + h];
    float yn = fmaf(gamma, (yv - mean) * rstd, beta);
    out[idx] = fmaf(resw[h], x[idx], gelu_t(yn));
}

// ---------------------------------------------------------------------------
extern "C" void kernel_launch(void* const* d_in, const int* in_sizes, int n_in,
                              void* d_out, int out_size, void* d_ws, size_t ws_size,
                              hipStream_t stream) {
    const float* x      = (const float*)d_in[0];
    const float* cond   = (const float*)d_in[1];
    const float* W_lin  = (const float*)d_in[2];
    const float* b_lin  = (const float*)d_in[3];
    const float* log_dt = (const float*)d_in[4];
    const float* A_re   = (const float*)d_in[5];
    const float* A_im   = (const float*)d_in[6];
    const float* C_re   = (const float*)d_in[7];
    const float* C_im   = (const float*)d_in[8];
    const float* Dv     = (const float*)d_in[9];
    const float* W_out  = (const float*)d_in[10];
    const float* b_out  = (const float*)d_in[11];
    const float* film_W = (const float*)d_in[12];
    const float* film_b = (const float*)d_in[13];
    const float* res_w  = (const float*)d_in[14];
    float* out = (float*)d_out;

    const size_t tot = (size_t)Bdim * Hdim * Ldim;
    float* ws   = (float*)d_ws;
    float* u    = ws;                                   // [B,H,L]
    float* v    = u + tot;                              // [B,H,L]
    float* yg   = u;                                    // alias: u dead after scan_out
    float* S    = v + tot;                              // [B*H*NCHUNK, N, 2]
    float* wr_  = S + (size_t)Bdim * Hdim * NCHUNK * 64;
    float* wi_  = wr_  + Hdim * Ndim;
    float* cdr_ = wi_  + Hdim * Ndim;
    float* cdi_ = cdr_ + Hdim * Ndim;
    float* wcr_ = cdi_ + Hdim * Ndim;
    float* wci_ = wcr_ + Hdim * Ndim;
    float* gb   = wci_ + Hdim * Ndim;                   // [B, 2H]
    float* stats = gb + Bdim * 2 * Hdim;                // [H, 2]

    k_setup<<<1, Hdim * Ndim, 0, stream>>>(log_dt, A_re, A_im, C_re, C_im,
                                           wr_, wi_, cdr_, cdi_, wcr_, wci_);
    k_film<<<1, Bdim * 2 * Hdim, 0, stream>>>(cond, film_W, film_b, gb);
    k_in_linear<<<(Bdim * (Ldim / 16)) / 8, 256, 0, stream>>>(x, W_lin, b_lin, u);
    k_scan_local<<<(Bdim * Hdim * NCHUNK) / 8, 256, 0, stream>>>(u, wr_, wi_, S);
    k_scan_prefix<<<(Bdim * Hdim * Ndim) / 256, 256, 0, stream>>>(S, wcr_, wci_);
    k_scan_out<<<(Bdim * Hdim * NCHUNK) / 8, 256, 0, stream>>>(u, S, wr_, wi_,
                                                               cdr_, cdi_, Dv, v);
    k_out_glu<<<(Bdim * (Ldim / 16)) / 8, 256, 0, stream>>>(v, W_out, b_out, yg);
    k_bn_stats<<<Hdim, 1024, 0, stream>>>(yg, stats);
    k_final<<<(int)(tot / 256), 256, 0, stream>>>(yg, stats, gb, res_w, x, out);
}